// RGCN_85899346255
// MI455X (gfx1250) — compile-verified
//
#include <hip/hip_runtime.h>
#include <hip/hip_bf16.h>

#define NU 30000
#define NI 15000
#define DD 64
#define RR 5
#define NU5 (NU * RR)
#define NI5 (NI * RR)
#define NADJ (NU + NI5)
#define NNZ_S 1200000
#define NNZ_A 840000

typedef __attribute__((ext_vector_type(2))) float v2f;
typedef __attribute__((ext_vector_type(8))) float v8f;

// ---------------------------------------------------------------------------
// SpMM scatter-add: 16 threads per nnz, each thread does a float4 gather and
// 4 native fp32 global atomics (unsafeAtomicAdd -> global_atomic_add_f32,
// no CAS loop). y must be pre-zeroed.
// ---------------------------------------------------------------------------
__global__ __launch_bounds__(256) void spmm_atomic_kernel(
    const int* __restrict__ rows, const int* __restrict__ cols,
    const float* __restrict__ vals, const float* __restrict__ x,
    float* __restrict__ y, int nnz) {
  int t = blockIdx.x * blockDim.x + threadIdx.x;
  int e = t >> 4;
  if (e >= nnz) return;
  int d = (t & 15) << 2;
  int r = rows[e];
  int c = cols[e];
  float v = vals[e];
  const float4 xv = *reinterpret_cast<const float4*>(x + (size_t)c * DD + d);
  float* yp = y + (size_t)r * DD + d;
  unsafeAtomicAdd(yp + 0, v * xv.x);
  unsafeAtomicAdd(yp + 1, v * xv.y);
  unsafeAtomicAdd(yp + 2, v * xv.z);
  unsafeAtomicAdd(yp + 3, v * xv.w);
}

// ---------------------------------------------------------------------------
// In-place PReLU, vectorized float4 (n must be a multiple of 4 — true for
// all uses here: multiples of 64).
// ---------------------------------------------------------------------------
__global__ __launch_bounds__(256) void prelu_kernel(
    float* __restrict__ p, size_t n4, const float* __restrict__ alpha) {
  size_t i = (size_t)blockIdx.x * blockDim.x + threadIdx.x;
  if (i >= n4) return;
  float a = alpha[0];
  float4 x = reinterpret_cast<float4*>(p)[i];
  x.x = (x.x >= 0.0f) ? x.x : a * x.x;
  x.y = (x.y >= 0.0f) ? x.y : a * x.y;
  x.z = (x.z >= 0.0f) ? x.z : a * x.z;
  x.w = (x.w >= 0.0f) ? x.w : a * x.w;
  reinterpret_cast<float4*>(p)[i] = x;
}

// ---------------------------------------------------------------------------
// Generic fp32 WMMA GEMM: Y[M x 64] = X[M x K] @ B (+ bias), where
//   wtrans == 0: B = W, W stored row-major K x 64
//   wtrans == 1: B = W^T, W stored row-major 64 x K   (B[k][n] = W[n*K + k])
// Block = 256 threads = 8 waves; block computes a 32-row x 64-col output tile,
// each wave one 16x16 tile via V_WMMA_F32_16X16X4_F32 chained over K.
// ---------------------------------------------------------------------------
__global__ __launch_bounds__(256) void gemm_wmma_kernel(
    const float* __restrict__ X, const float* __restrict__ W,
    float* __restrict__ Y, int M, int K, int wtrans,
    const float* __restrict__ bias) {
  __shared__ float Xs[32][129];   // padded: conflict-free column reads
  __shared__ float Ws[128][65];   // padded: conflict-free transposed writes

  const int tid = threadIdx.x;
  const int m0 = blockIdx.x * 32;

  // Stage weights into LDS (coalesced global reads either way).
  if (wtrans == 0) {
    for (int i = tid; i < K * 64; i += 256) {
      int k = i >> 6;
      int n = i & 63;
      Ws[k][n] = W[i];
    }
  } else {
    for (int i = tid; i < K * 64; i += 256) {
      int n = i / K;
      int k = i - n * K;
      Ws[k][n] = W[i];
    }
  }
  // Stage 32 x K tile of X (zero-padded past M).
  for (int i = tid; i < 32 * K; i += 256) {
    int r = i / K;
    int k = i - r * K;
    int m = m0 + r;
    Xs[r][k] = (m < M) ? X[(size_t)m * K + k] : 0.0f;
  }
  __syncthreads();

  const int wave  = tid >> 5;
  const int lane  = tid & 31;
  const int hi    = lane >> 4;   // 0: lanes 0-15, 1: lanes 16-31
  const int l16   = lane & 15;
  const int rbase = (wave >> 2) * 16;  // row tile within block (0 or 16)
  const int n0    = (wave & 3) * 16;   // col tile (0,16,32,48)

  v8f c = {};
  for (int k = 0; k < K; k += 4) {
    v2f a, b;
    // A 16x4 f32 layout: VGPR0 = K (lanes0-15) / K+2 (lanes16-31); VGPR1 = +1
    a.x = Xs[rbase + l16][k + 2 * hi];
    a.y = Xs[rbase + l16][k + 2 * hi + 1];
    // B 4x16 f32 layout mirrors A: rows striped across lanes within VGPR
    b.x = Ws[k + 2 * hi][n0 + l16];
    b.y = Ws[k + 2 * hi + 1][n0 + l16];
    c = __builtin_amdgcn_wmma_f32_16x16x4_f32(
        /*neg_a=*/false, a, /*neg_b=*/false, b,
        /*c_mod=*/(short)0, c, /*reuse_a=*/false, /*reuse_b=*/false);
  }

  float bv = bias ? bias[n0 + l16] : 0.0f;
  // C/D layout: VGPR i -> M = i (lanes 0-15) or i+8 (lanes 16-31), N = l16
#pragma unroll
  for (int i = 0; i < 8; ++i) {
    int m = m0 + rbase + i + 8 * hi;
    if (m < M) Y[(size_t)m * 64 + n0 + l16] = c[i] + bv;
  }
}

// ---------------------------------------------------------------------------
// u_social[u][c] = sum_r concat(e0,e1)[u*5+r][c] / 4,  c in [0,128)
// ---------------------------------------------------------------------------
__global__ __launch_bounds__(256) void usocial_kernel(
    const float* __restrict__ e0, const float* __restrict__ e1,
    float* __restrict__ out) {
  int t = blockIdx.x * blockDim.x + threadIdx.x;
  if (t >= NU * 128) return;
  int u = t >> 7;
  int c = t & 127;
  const float* src = (c < 64) ? e0 : e1;
  int cc = c & 63;
  float s = 0.0f;
#pragma unroll
  for (int r = 0; r < RR; ++r) s += src[(size_t)(u * RR + r) * 64 + cc];
  out[t] = s * 0.25f;
}

// ---------------------------------------------------------------------------
// Final outputs:
//  user_embedding[u][c] = (c<64) ? emb[u][c] : emb1[u][c-64]
//  item_embedding[i][c] = sum_r concat(ai0,ai1)[i*5+r][c] / 5
// d_out = user_embedding (NU*128) ++ item_embedding (NI*128)
// ---------------------------------------------------------------------------
__global__ __launch_bounds__(256) void finalize_kernel(
    const float* __restrict__ emb, const float* __restrict__ emb1,
    float* __restrict__ out) {
  int t = blockIdx.x * blockDim.x + threadIdx.x;
  if (t >= NU * 128 + NI * 128) return;
  if (t < NU * 128) {
    int u = t >> 7;
    int c = t & 127;
    out[t] = (c < 64) ? emb[(size_t)u * 64 + c]
                      : emb1[(size_t)u * 64 + (c - 64)];
  } else {
    int t2 = t - NU * 128;
    int i = t2 >> 7;
    int c = t2 & 127;
    const float* src = (c < 64) ? emb : emb1;
    int cc = c & 63;
    float s = 0.0f;
#pragma unroll
    for (int r = 0; r < RR; ++r)
      s += src[(size_t)(NU + i * RR + r) * 64 + cc];
    out[t] = s * 0.2f;
  }
}

// ---------------------------------------------------------------------------
// Orchestration
// ---------------------------------------------------------------------------
extern "C" void kernel_launch(void* const* d_in, const int* in_sizes, int n_in,
                              void* d_out, int out_size, void* d_ws, size_t ws_size,
                              hipStream_t stream) {
  const float* user_emb    = (const float*)d_in[0];
  const float* item_emb    = (const float*)d_in[1];
  const float* w_r1_w      = (const float*)d_in[2];   // 64 x 128 row-major
  const float* w_r1_b      = (const float*)d_in[3];   // 64
  const float* social_w1   = (const float*)d_in[4];   // 64 x 64
  const float* user_w1     = (const float*)d_in[5];   // 64 x 64
  const float* item_w1     = (const float*)d_in[6];   // 64 x 64
  const float* alpha       = (const float*)d_in[7];   // 1
  const float* social_vals = (const float*)d_in[8];
  const float* adj_vals    = (const float*)d_in[9];
  const int*   social_rows = (const int*)d_in[10];
  const int*   social_cols = (const int*)d_in[11];
  const int*   adj_rows    = (const int*)d_in[12];
  const int*   adj_cols    = (const int*)d_in[13];
  float* out = (float*)d_out;

  // Three reusable fp32 buffers of NU5*64 each (115.2 MB total).
  const size_t BUF = (size_t)NU5 * 64;
  float* B1 = (float*)d_ws;        // e0 -> ego -> emb1
  float* B2 = B1 + BUF;            // t  -> u_social -> ego1
  float* B3 = B2 + BUF;            // e1 -> emb

  const int TB = 256;
  const int spmm_s_grid = (NNZ_S * 16 + TB - 1) / TB;
  const int spmm_a_grid = (NNZ_A * 16 + TB - 1) / TB;
  const size_t NADJF = (size_t)NADJ * 64;

  // ---- e0 = prelu(spmm(social, user_emb)) -> B1
  hipMemsetAsync(B1, 0, BUF * sizeof(float), stream);
  spmm_atomic_kernel<<<spmm_s_grid, TB, 0, stream>>>(
      social_rows, social_cols, social_vals, user_emb, B1, NNZ_S);
  prelu_kernel<<<(int)((BUF / 4 + TB - 1) / TB), TB, 0, stream>>>(B1, BUF / 4, alpha);

  // ---- t = e0 @ social_w1 -> B2
  gemm_wmma_kernel<<<(NU5 + 31) / 32, TB, 0, stream>>>(
      B1, social_w1, B2, NU5, 64, 0, nullptr);

  // ---- e1 = prelu(spmm(social, t)) -> B3
  hipMemsetAsync(B3, 0, BUF * sizeof(float), stream);
  spmm_atomic_kernel<<<spmm_s_grid, TB, 0, stream>>>(
      social_rows, social_cols, social_vals, B2, B3, NNZ_S);
  prelu_kernel<<<(int)((BUF / 4 + TB - 1) / TB), TB, 0, stream>>>(B3, BUF / 4, alpha);

  // ---- u_social = reduce(concat(e0,e1)) / 4 -> B2 (t is dead)
  usocial_kernel<<<(NU * 128 + TB - 1) / TB, TB, 0, stream>>>(B1, B3, B2);

  // ---- u = u_social @ w_r1_w.T + b -> ego rows [0,NU) in B1 (e0 is dead)
  gemm_wmma_kernel<<<(NU + 31) / 32, TB, 0, stream>>>(
      B2, w_r1_w, B1, NU, 128, 1, w_r1_b);

  // ---- ego rows [NU, NADJ) = item_emb
  hipMemcpyAsync(B1 + (size_t)NU * 64, item_emb,
                 (size_t)NI5 * 64 * sizeof(float),
                 hipMemcpyDeviceToDevice, stream);

  // ---- emb = prelu(spmm(adj, ego)) -> B3 (e1 is dead)
  hipMemsetAsync(B3, 0, NADJF * sizeof(float), stream);
  spmm_atomic_kernel<<<spmm_a_grid, TB, 0, stream>>>(
      adj_rows, adj_cols, adj_vals, B1, B3, NNZ_A);
  prelu_kernel<<<(int)((NADJF / 4 + TB - 1) / TB), TB, 0, stream>>>(B3, NADJF / 4, alpha);

  // ---- ego1 = concat(au0 @ user_w1, ai0 @ item_w1) -> B2 (u_social is dead)
  gemm_wmma_kernel<<<(NU + 31) / 32, TB, 0, stream>>>(
      B3, user_w1, B2, NU, 64, 0, nullptr);
  gemm_wmma_kernel<<<(NI5 + 31) / 32, TB, 0, stream>>>(
      B3 + (size_t)NU * 64, item_w1, B2 + (size_t)NU * 64, NI5, 64, 0, nullptr);

  // ---- emb1 = prelu(spmm(adj, ego1)) -> B1 (ego is dead)
  hipMemsetAsync(B1, 0, NADJF * sizeof(float), stream);
  spmm_atomic_kernel<<<spmm_a_grid, TB, 0, stream>>>(
      adj_rows, adj_cols, adj_vals, B2, B1, NNZ_A);
  prelu_kernel<<<(int)((NADJF / 4 + TB - 1) / TB), TB, 0, stream>>>(B1, NADJF / 4, alpha);

  // ---- outputs
  finalize_kernel<<<(NU * 128 + NI * 128 + TB - 1) / TB, TB, 0, stream>>>(
      B3, B1, out);
}